// MemoryL2EmbeddingLoss_65335042506955
// MI455X (gfx1250) — compile-verified
//
#include <hip/hip_runtime.h>

#define B_ROWS 1024
#define D_DIM  512
#define M_COLS 32768

typedef float v2f __attribute__((ext_vector_type(2)));
typedef float v8f __attribute__((ext_vector_type(8)));

// ---------------------------------------------------------------- zero scratch
__global__ __launch_bounds__(256) void zero_acc_kernel(float* __restrict__ rowacc) {
    int i = blockIdx.x * 256 + threadIdx.x;
    if (i < 4 * B_ROWS) rowacc[i] = 0.0f;
}

// ------------------------------------------------- squared norms of all refs
// ref row j = embeddings[j] for j<B, else emb_mem[j-B]. One wave per row.
__global__ __launch_bounds__(256) void norms_kernel(const float* __restrict__ emb,
                                                    const float* __restrict__ mem,
                                                    float* __restrict__ sq_ref) {
    const int row  = blockIdx.x * 8 + (threadIdx.x >> 5);
    const int lane = threadIdx.x & 31;
    if (row >= M_COLS) return;
    const float* src = (row < B_ROWS) ? (emb + (size_t)row * D_DIM)
                                      : (mem + (size_t)(row - B_ROWS) * D_DIM);
    float s = 0.0f;
#pragma unroll
    for (int it = 0; it < D_DIM / 128; ++it) {
        const float4 v = *(const float4*)(src + it * 128 + lane * 4);
        s += v.x * v.x + v.y * v.y + v.z * v.z + v.w * v.w;
    }
#pragma unroll
    for (int off = 16; off; off >>= 1) s += __shfl_xor(s, off, 32);
    if (lane == 0) sq_ref[row] = s;
}

// ------------------------------------------------------- main fused GEMM+loss
// Grid: (M/256, B/128). Block: 256 threads = 8 waves.
// Wave (wm in 0..1, wn in 0..3) owns a 64x64 patch = 4x4 WMMA 16x16 fragments:
// 8 b64 loads feed 16 v_wmma_f32_16x16x4_f32 per K-step (0.5 loads/WMMA).
__global__ __launch_bounds__(256) void tile_loss_kernel(
        const float* __restrict__ emb,   const int* __restrict__ labels,
        const float* __restrict__ mem,   const int* __restrict__ lbl_mem,
        const float* __restrict__ sq_ref, float* __restrict__ rowacc) {
    __shared__ float s_sp[128], s_cp[128], s_sn[128], s_cn[128];

    const int tid  = threadIdx.x;
    const int lane = tid & 31;
    const int wave = tid >> 5;
    const int wm   = wave & 1;     // row half of WG tile
    const int wn   = wave >> 1;    // col quarter of WG tile
    const int half = lane >> 4;    // K sub-lane group (0/1)
    const int l16  = lane & 15;    // row/col within fragment

    for (int t = tid; t < 128; t += 256) { s_sp[t] = 0.f; s_cp[t] = 0.f; s_sn[t] = 0.f; s_cn[t] = 0.f; }
    __syncthreads();

    const int  rowBase    = blockIdx.y * 128 + wm * 64;
    const int  colWG      = blockIdx.x * 256;
    const bool colInBatch = (colWG < B_ROWS);          // 1024 % 256 == 0: tile is on one side
    const float* colsrc   = colInBatch ? (emb + (size_t)colWG * D_DIM)
                                       : (mem + (size_t)(colWG - B_ROWS) * D_DIM);

    // 32-bit A/B fragment layout: lanes 0-15 hold K = 0,1 ; lanes 16-31 hold K = 2,3
    const float* ap[4];
    const float* bp[4];
#pragma unroll
    for (int i = 0; i < 4; ++i) {
        ap[i] = emb    + (size_t)(rowBase + i * 16 + l16)  * D_DIM + half * 2;
        bp[i] = colsrc + (size_t)(wn * 64 + i * 16 + l16)  * D_DIM + half * 2;
    }

    v8f acc[4][4];
#pragma unroll
    for (int i = 0; i < 4; ++i)
#pragma unroll
        for (int j = 0; j < 4; ++j)
#pragma unroll
            for (int r = 0; r < 8; ++r) acc[i][j][r] = 0.0f;

#pragma clang loop unroll(disable)
    for (int kc = 0; kc < D_DIM; kc += 32) {
#pragma unroll
        for (int i = 0; i < 4; ++i) {   // pull data ~4 chunks ahead toward the WGP
            __builtin_prefetch(ap[i] + kc + 128, 0, 0);
            __builtin_prefetch(bp[i] + kc + 128, 0, 0);
        }
#pragma unroll
        for (int k = kc; k < kc + 32; k += 4) {
            v2f a[4], b[4];
#pragma unroll
            for (int i = 0; i < 4; ++i) {
                a[i] = *(const v2f*)(ap[i] + k);
                b[i] = *(const v2f*)(bp[i] + k);
            }
#pragma unroll
            for (int i = 0; i < 4; ++i)
#pragma unroll
                for (int j = 0; j < 4; ++j)
                    acc[i][j] = __builtin_amdgcn_wmma_f32_16x16x4_f32(
                        false, a[i], false, b[j], (short)0, acc[i][j], false, false);
        }
    }

    // Epilogue: C/D layout (VGPR r -> M = r + 8*half, N = l16) -> (m, n);
    // apply margins + masks; reduce into LDS (branches rarely taken for this data).
#pragma unroll
    for (int j = 0; j < 4; ++j) {
        const int   n    = colWG + wn * 64 + j * 16 + l16;
        const float sqb  = sq_ref[n];
        const int   labn = colInBatch ? labels[n] : lbl_mem[n - B_ROWS];
#pragma unroll
        for (int i = 0; i < 4; ++i) {
            const int m0 = rowBase + i * 16 + 8 * half;
#pragma unroll
            for (int r = 0; r < 8; ++r) {
                const int m = m0 + r;
                float dd = sq_ref[m] + sqb - 2.0f * acc[i][j][r];
                dd = fmaxf(dd, 0.0f);
                if (n != m) {                          // not_self
                    const int lm = wm * 64 + i * 16 + 8 * half + r;   // local row 0..127
                    if (labn == labels[m]) {           // same-label: pos branch
                        if (dd > 0.0f) { atomicAdd(&s_sp[lm], dd); atomicAdd(&s_cp[lm], 1.0f); }
                    } else {                           // diff-label: neg branch
                        const float ln = 1.0f - dd;
                        if (ln > 0.0f) { atomicAdd(&s_sn[lm], ln); atomicAdd(&s_cn[lm], 1.0f); }
                    }
                }
            }
        }
    }
    __syncthreads();

    if (tid < 128) {
        const int row = blockIdx.y * 128 + tid;
        unsafeAtomicAdd(&rowacc[row],              s_sp[tid]);
        unsafeAtomicAdd(&rowacc[B_ROWS + row],     s_cp[tid]);
        unsafeAtomicAdd(&rowacc[2 * B_ROWS + row], s_sn[tid]);
        unsafeAtomicAdd(&rowacc[3 * B_ROWS + row], s_cn[tid]);
    }
}

// ----------------------------------------------------------------- finalize
__global__ __launch_bounds__(256) void finalize_kernel(const float* __restrict__ rowacc,
                                                       float* __restrict__ out) {
    __shared__ float red[256];
    float s = 0.0f;
    for (int row = threadIdx.x; row < B_ROWS; row += 256) {
        const float sp = rowacc[row];
        const float cp = rowacc[B_ROWS + row];
        const float sn = rowacc[2 * B_ROWS + row];
        const float cn = rowacc[3 * B_ROWS + row];
        s += sp / (cp + 1e-6f) + sn / (cn + 1e-6f);
    }
    red[threadIdx.x] = s;
    __syncthreads();
    for (int off = 128; off; off >>= 1) {
        if (threadIdx.x < off) red[threadIdx.x] += red[threadIdx.x + off];
        __syncthreads();
    }
    if (threadIdx.x == 0) out[0] = red[0] / (float)B_ROWS;
}

// ----------------------------------------------------------------- launcher
extern "C" void kernel_launch(void* const* d_in, const int* in_sizes, int n_in,
                              void* d_out, int out_size, void* d_ws, size_t ws_size,
                              hipStream_t stream) {
    const float* emb     = (const float*)d_in[0];   // [1024, 512]
    const int*   labels  = (const int*)d_in[1];     // [1024]
    const float* mem     = (const float*)d_in[2];   // [31744, 512]
    const int*   lbl_mem = (const int*)d_in[3];     // [31744]
    // d_in[4] (add_to_mem) does not affect the returned loss
    float* out = (float*)d_out;

    float* sq_ref = (float*)d_ws;          // M_COLS floats
    float* rowacc = sq_ref + M_COLS;       // 4*B_ROWS floats (sp, cp, sn, cn)

    zero_acc_kernel<<<(4 * B_ROWS + 255) / 256, 256, 0, stream>>>(rowacc);
    norms_kernel<<<M_COLS / 8, 256, 0, stream>>>(emb, mem, sq_ref);
    dim3 grid(M_COLS / 256, B_ROWS / 128);
    tile_loss_kernel<<<grid, 256, 0, stream>>>(emb, labels, mem, lbl_mem, sq_ref, rowacc);
    finalize_kernel<<<1, 256, 0, stream>>>(rowacc, out);
}